// Dense2Det_39204461478180
// MI455X (gfx1250) — compile-verified
//
#include <hip/hip_runtime.h>
#include <hip/hip_bf16.h>

#ifndef __has_builtin
#define __has_builtin(x) 0
#endif

typedef float v2f __attribute__((ext_vector_type(2)));
typedef float v8f __attribute__((ext_vector_type(8)));

#define NIMG         2
#define M_TOT        8960      // slots per image; top-k fills 6960 (2000+2000+2000+768+192)
#define M_PAD        16384     // bitonic padding
#define NWORDS       280       // 8960 / 32
#define MAX_OUT      1000
#define NMS_THRF     0.7f
#define MAX_RATIO_F  4.135166556742356f   // |log(16/1000)|
#define SCORE_STRIDE 65536
#define SCORE_TOTAL  65472     // 49152+12288+3072+768+192

// ---------- helpers ----------
__device__ __forceinline__ unsigned f2ord(float f) {
  unsigned u = __float_as_uint(f);
  return (u & 0x80000000u) ? ~u : (u | 0x80000000u);   // monotonic float->uint
}

__device__ __forceinline__ unsigned ballot32(bool p) {
#if __has_builtin(__builtin_amdgcn_ballot_w32)
  return __builtin_amdgcn_ballot_w32(p);               // wave32-native
#else
  return (unsigned)__ballot((int)p);
#endif
}

__device__ __forceinline__ unsigned pick8(const unsigned b[8], int q) {
  unsigned r = b[0];
  r = (q == 1) ? b[1] : r; r = (q == 2) ? b[2] : r; r = (q == 3) ? b[3] : r;
  r = (q == 4) ? b[4] : r; r = (q == 5) ? b[5] : r; r = (q == 6) ? b[6] : r;
  r = (q == 7) ? b[7] : r;
  return r;
}

// ---------- 0: zero d_out ----------
__global__ void zero_kernel(float* out, int n) {
  int i = blockIdx.x * blockDim.x + threadIdx.x;
  if (i < n) out[i] = 0.0f;
}

// ---------- 1: sigmoid scores, HWC order ----------
__global__ void score_kernel(const float* c0, const float* c1, const float* c2,
                             const float* c3, const float* c4, float* scores) {
  int tid = blockIdx.x * blockDim.x + threadIdx.x;
  if (tid >= NIMG * SCORE_TOTAL) return;
  int n = tid / SCORE_TOTAL, r = tid % SCORE_TOTAL;
  int off, H; const float* cp;
  if      (r < 49152) { off = 0;     H = 128; cp = c0; }
  else if (r < 61440) { off = 49152; H = 64;  cp = c1; }
  else if (r < 64512) { off = 61440; H = 32;  cp = c2; }
  else if (r < 65280) { off = 64512; H = 16;  cp = c3; }
  else                { off = 65280; H = 8;   cp = c4; }
  int idx = r - off;
  int a = idx % 3, pos = idx / 3, w = pos % H, h = pos / H;
  float x = cp[((n * 3 + a) * H + h) * H + w];
  scores[n * SCORE_STRIDE + off + idx] = 1.0f / (1.0f + expf(-x));
}

// ---------- 2: per-(image,level) top-k via 32-step threshold binary search ----------
__global__ __launch_bounds__(1024) void topk_kernel(const float* scores, int* sel) {
  const int Ls[5]    = {49152, 12288, 3072, 768, 192};
  const int offs[5]  = {0, 49152, 61440, 64512, 65280};
  const int ks[5]    = {2000, 2000, 2000, 768, 192};
  const int bases[5] = {0, 2000, 4000, 6000, 6768};
  int n = blockIdx.x / 5, lvl = blockIdx.x % 5;
  int L = Ls[lvl], k = ks[lvl];
  const float* sp = scores + n * SCORE_STRIDE + offs[lvl];
  int* outp = sel + n * M_TOT + bases[lvl];
  int tid = threadIdx.x, NT = blockDim.x;

  __shared__ unsigned slo, shi;
  __shared__ int scnt, pg, pe;
  if (tid == 0) { slo = 0u; shi = 0xFFFFFFFFu; }
  __syncthreads();
  for (int it = 0; it < 32; ++it) {             // invariant: cnt(>=lo)>=k, cnt(>=hi)<k
    unsigned lo = slo, hi = shi;
    if (hi - lo <= 1u) break;                   // uniform
    unsigned mid = lo + ((hi - lo) >> 1);
    if (tid == 0) scnt = 0;
    __syncthreads();
    int c = 0;
    for (int i = tid; i < L; i += NT) c += (__float_as_uint(sp[i]) >= mid);
    atomicAdd(&scnt, c);
    __syncthreads();
    if (tid == 0) { if (scnt >= k) slo = mid; else shi = mid; }
    __syncthreads();
  }
  unsigned T = slo;
  if (tid == 0) { scnt = 0; pg = 0; pe = 0; }
  __syncthreads();
  int cg = 0;
  for (int i = tid; i < L; i += NT) cg += (__float_as_uint(sp[i]) > T);
  atomicAdd(&scnt, cg);
  __syncthreads();
  int cntGreater = scnt, extra = k - cntGreater;
  for (int i = tid; i < L; i += NT) {
    unsigned u = __float_as_uint(sp[i]);
    if (u > T)       { int p = atomicAdd(&pg, 1); outp[p] = i; }
    else if (u == T) { int p = atomicAdd(&pe, 1); if (p < extra) outp[cntGreater + p] = i; }
  }
}

// ---------- 3: gather + anchor + decode + sort keys ----------
__global__ void decode_kernel(const float* r0, const float* r1, const float* r2,
                              const float* r3, const float* r4,
                              const float* scores, const int* sel,
                              const int* ph, const int* pw,
                              float* boxDec, float* scDec, unsigned long long* keys) {
  int tid = blockIdx.x * blockDim.x + threadIdx.x;
  if (tid >= NIMG * M_PAD) return;
  int n = tid / M_PAD, m = tid % M_PAD;
  unsigned long long* kp = keys + (size_t)n * M_PAD;
  if (m >= M_TOT) { kp[m] = ~0ull; return; }

  int off, H, stride; const float* rp;
  if      (m < 2000) { off = 0;     H = 128; stride = 8;   rp = r0; }
  else if (m < 4000) { off = 49152; H = 64;  stride = 16;  rp = r1; }
  else if (m < 6000) { off = 61440; H = 32;  stride = 32;  rp = r2; }
  else if (m < 6768) { off = 64512; H = 16;  stride = 64;  rp = r3; }
  else               { off = 65280; H = 8;   stride = 128; rp = r4; }

  int idx = sel[n * M_TOT + m];
  int a = idx % 3, pos = idx / 3, w = pos % H, h = pos / H;
  float s = scores[n * SCORE_STRIDE + off + idx];

  float dx = rp[((n * 12 + a * 4 + 0) * H + h) * H + w];
  float dy = rp[((n * 12 + a * 4 + 1) * H + h) * H + w];
  float dw = rp[((n * 12 + a * 4 + 2) * H + h) * H + w];
  float dh = rp[((n * 12 + a * 4 + 3) * H + h) * H + w];
  dw = fminf(fmaxf(dw, -MAX_RATIO_F), MAX_RATIO_F);
  dh = fminf(fmaxf(dh, -MAX_RATIO_F), MAX_RATIO_F);

  float b8 = 8.0f * (float)stride, sw, sh;     // ratios {0.5,1,2}
  if      (a == 0) { sw = b8 * 1.41421356237f; sh = b8 * 0.70710678119f; }
  else if (a == 1) { sw = b8;                  sh = b8; }
  else             { sw = b8 * 0.70710678119f; sh = b8 * 1.41421356237f; }

  float px = (float)w * (float)stride, py = (float)h * (float)stride;
  float gx = px + sw * dx, gy = py + sh * dy;
  float gw = sw * expf(dw), gh = sh * expf(dh);
  float W = (float)pw[0], Hc = (float)ph[0];
  float x1 = fminf(fmaxf(gx - 0.5f * gw, 0.0f), W);
  float y1 = fminf(fmaxf(gy - 0.5f * gh, 0.0f), Hc);
  float x2 = fminf(fmaxf(gx + 0.5f * gw, 0.0f), W);
  float y2 = fminf(fmaxf(gy + 0.5f * gh, 0.0f), Hc);
  bool valid = (x2 - x1 > 0.0f) && (y2 - y1 > 0.0f);
  float su = valid ? s : -1.0f;

  int g = n * M_TOT + m;
  boxDec[4 * g + 0] = x1; boxDec[4 * g + 1] = y1;
  boxDec[4 * g + 2] = x2; boxDec[4 * g + 3] = y2;
  scDec[g] = su;
  kp[m] = ((unsigned long long)(~f2ord(su)) << 32) | (unsigned)m;  // ascending == score desc, tie->low idx
}

// ---------- 4: bitonic sort (ascending u64), one (k,j) pass per launch ----------
__global__ void bitonic_kernel(unsigned long long* keys, int k, int j) {
  int tid = blockIdx.x * blockDim.x + threadIdx.x;
  int n = tid >> 14, i = tid & (M_PAD - 1);
  int l = i ^ j;
  if (l > i) {
    unsigned long long* kp = keys + (size_t)n * M_PAD;
    unsigned long long av = kp[i], bv = kp[l];
    bool up = ((i & k) == 0);
    if ((av > bv) == up) { kp[i] = bv; kp[l] = av; }
  }
}

// ---------- 5: permute boxes/scores into sorted order ----------
__global__ void reorder_kernel(const unsigned long long* keys, const float* boxDec,
                               const float* scDec, float* boxS, float* scS) {
  int tid = blockIdx.x * blockDim.x + threadIdx.x;
  if (tid >= NIMG * M_TOT) return;
  int n = tid / M_TOT, i = tid % M_TOT;
  unsigned src = (unsigned)(keys[(size_t)n * M_PAD + i] & 0xFFFFFFFFu);
  int g = n * M_TOT + i, gs = n * M_TOT + (int)src;
  boxS[4 * g + 0] = boxDec[4 * gs + 0]; boxS[4 * g + 1] = boxDec[4 * gs + 1];
  boxS[4 * g + 2] = boxDec[4 * gs + 2]; boxS[4 * g + 3] = boxDec[4 * gs + 3];
  scS[g] = scDec[gs];
}

// ---------- 6: suppression bitmask. One wave per 16x32 tile.
// WMMA f32 16x16x4 computes the pairwise center-distance^2 tile:
//   A[m][*] = [cx_i, cy_i, |c_i|^2, 1],  B[*][n] = [-2cx_j, -2cy_j, 1, |c_j|^2]^T
//   D[m][n] = |c_i - c_j|^2.  dist^2 > 2(r_i^2+r_j^2) >= (r_i+r_j)^2  =>  IoU == 0.
// IoU threshold test is division-free: inter/max(uni,eps) > t  <=>  inter > t*max(uni,eps).
__global__ __launch_bounds__(128) void mask_kernel(const float* boxS, unsigned* mask) {
  const int jobsPerImg = (M_TOT / 16) * NWORDS;            // 560*280
  int lane = threadIdx.x & 31;
  long long job = (long long)blockIdx.x * 4 + (threadIdx.x >> 5);
  int n  = (int)(job / jobsPerImg);
  int rm = (int)(job % jobsPerImg);
  int tr = rm / NWORDS, cw = rm % NWORDS;
  int r0 = tr * 16, c0 = cw * 32;
  const float* bp = boxS + (size_t)n * M_TOT * 4;

  int  m16 = lane & 15;
  bool hiH = lane >= 16;
  // A operand: 16x4 f32, lanes 0-15 hold K=0..1, lanes 16-31 hold K=2..3 (per ISA layout)
  float rx1 = bp[4 * (r0 + m16) + 0], ry1 = bp[4 * (r0 + m16) + 1];
  float rx2 = bp[4 * (r0 + m16) + 2], ry2 = bp[4 * (r0 + m16) + 3];
  float cxr = 0.5f * (rx1 + rx2), cyr = 0.5f * (ry1 + ry2);
  float sqr = cxr * cxr + cyr * cyr;
  v2f A; A.x = hiH ? sqr : cxr; A.y = hiH ? 1.0f : cyr;

  unsigned bal[2][8];
#pragma unroll
  for (int t = 0; t < 2; ++t) {
    int jc = c0 + t * 16 + m16;                            // this lane's column box
    float jx1 = bp[4 * jc + 0], jy1 = bp[4 * jc + 1];
    float jx2 = bp[4 * jc + 2], jy2 = bp[4 * jc + 3];
    float cxc = 0.5f * (jx1 + jx2), cyc = 0.5f * (jy1 + jy2);
    float sqc = cxc * cxc + cyc * cyc;
    v2f B; B.x = hiH ? 1.0f : (-2.0f * cxc); B.y = hiH ? sqc : (-2.0f * cyc);
    v8f C = {};
#if __has_builtin(__builtin_amdgcn_wmma_f32_16x16x4_f32)
    v8f D = __builtin_amdgcn_wmma_f32_16x16x4_f32(false, A, false, B, (short)0, C,
                                                  false, false);
#else
    v8f D;                                                 // scalar fallback (non-gfx1250)
#pragma unroll
    for (int v = 0; v < 8; ++v) {
      int mr = v + (hiH ? 8 : 0);
      float ax = 0.5f * (bp[4 * (r0 + mr) + 0] + bp[4 * (r0 + mr) + 2]) - cxc;
      float ay = 0.5f * (bp[4 * (r0 + mr) + 1] + bp[4 * (r0 + mr) + 3]) - cyc;
      D[v] = ax * ax + ay * ay;
    }
#endif
    float aj  = (jx2 - jx1) * (jy2 - jy1);
    float rc2 = 0.25f * ((jx2 - jx1) * (jx2 - jx1) + (jy2 - jy1) * (jy2 - jy1));
#pragma unroll
    for (int v = 0; v < 8; ++v) {                          // D lane map: n=lane&15, m=v(+8 hi)
      int mr = v + (hiH ? 8 : 0);
      int gi = r0 + mr;
      float ix1 = bp[4 * gi + 0], iy1 = bp[4 * gi + 1];
      float ix2 = bp[4 * gi + 2], iy2 = bp[4 * gi + 3];
      float rr2 = 0.25f * ((ix2 - ix1) * (ix2 - ix1) + (iy2 - iy1) * (iy2 - iy1));
      bool near = (jc > gi) && (D[v] <= 2.0f * (rr2 + rc2) + 8.0f);
      float ai = (ix2 - ix1) * (iy2 - iy1);
      float iw = fmaxf(fminf(ix2, jx2) - fmaxf(ix1, jx1), 0.0f);
      float ih = fmaxf(fminf(iy2, jy2) - fmaxf(iy1, jy1), 0.0f);
      float inter = iw * ih;
      // division-free IoU test: denominator is strictly positive
      bool over = inter > NMS_THRF * fmaxf(ai + aj - inter, 1e-9f);
      bal[t][v] = ballot32(near && over);                  // all lanes execute
    }
  }
  // ballot bit L: L<16 -> (m=v, n=L);  L>=16 -> (m=v+8, n=L-16). Assemble 32-col words.
  int q = lane & 7;
  unsigned s0 = pick8(bal[0], q), s1 = pick8(bal[1], q);
  unsigned word = (lane < 8) ? ((s0 & 0xFFFFu) | ((s1 & 0xFFFFu) << 16))
                             : ((s0 >> 16) | (s1 & 0xFFFF0000u));
  if (lane < 16)
    mask[((size_t)n * M_TOT + (r0 + lane)) * NWORDS + cw] = word;
}

// ---------- 7: sequential greedy scan over LDS bitset ----------
__global__ __launch_bounds__(512) void nms_scan_kernel(const float* scS, const unsigned* mask,
                                                       int* keep) {
  int n = blockIdx.x, tid = threadIdx.x;
  __shared__ unsigned remv[NWORDS];
  __shared__ int curKeep;
  for (int i = tid; i < NWORDS; i += blockDim.x) remv[i] = 0u;
  __syncthreads();
  const float* sp = scS + n * M_TOT;
  const unsigned* mp = mask + (size_t)n * M_TOT * NWORDS;
  for (int i = 0; i < M_TOT; ++i) {
    if (tid == 0)
      curKeep = (sp[i] > 0.0f) && !((remv[i >> 5] >> (i & 31)) & 1u);
    __syncthreads();
    if (curKeep && tid < NWORDS) remv[tid] |= mp[(size_t)i * NWORDS + tid];
    __syncthreads();
  }
  for (int i = tid; i < M_TOT; i += blockDim.x)
    keep[n * M_TOT + i] = (sp[i] > 0.0f) && !((remv[i >> 5] >> (i & 31)) & 1u);
}

// ---------- 8: ordered compaction -> padded outputs ----------
__global__ __launch_bounds__(1024) void output_kernel(const float* boxS, const float* scS,
                                                      const int* keep, float* out) {
  const int CH = 9;                                        // 1024*9 >= 8960
  int n = blockIdx.x, tid = threadIdx.x;
  __shared__ unsigned sc[1024];
  int start = tid * CH;
  unsigned c = 0;
  for (int u = 0; u < CH; ++u) { int i = start + u; c += (i < M_TOT && keep[n * M_TOT + i]); }
  sc[tid] = c;
  __syncthreads();
  for (int d = 1; d < 1024; d <<= 1) {
    unsigned v = (tid >= d) ? sc[tid - d] : 0u;
    __syncthreads();
    sc[tid] += v;
    __syncthreads();
  }
  unsigned pos = sc[tid] - c;                              // exclusive prefix
  for (int u = 0; u < CH; ++u) {
    int i = start + u;
    if (i < M_TOT && keep[n * M_TOT + i]) {
      if (pos < MAX_OUT) {
        int g = n * M_TOT + i, o = n * MAX_OUT + (int)pos;
        out[4 * o + 0] = boxS[4 * g + 0]; out[4 * o + 1] = boxS[4 * g + 1];
        out[4 * o + 2] = boxS[4 * g + 2]; out[4 * o + 3] = boxS[4 * g + 3];
        out[NIMG * MAX_OUT * 4 + o] = scS[g];              // scores
        // labels (offset 5*NIMG*MAX_OUT..) stay zero
        out[NIMG * MAX_OUT * 6 + o] = 1.0f;                // ok flag
      }
      pos++;
    }
  }
}

// ---------- host ----------
extern "C" void kernel_launch(void* const* d_in, const int* in_sizes, int n_in,
                              void* d_out, int out_size, void* d_ws, size_t ws_size,
                              hipStream_t stream) {
  const float* cls[5]; const float* reg[5];
  for (int i = 0; i < 5; ++i) { cls[i] = (const float*)d_in[i]; reg[i] = (const float*)d_in[5 + i]; }
  const int* img_h = (const int*)d_in[10];
  const int* img_w = (const int*)d_in[11];
  float* out = (float*)d_out;

  size_t off = 0;
  auto take = [&](size_t bytes) { size_t r = off; off += (bytes + 255) & ~(size_t)255; return r; };
  char* ws = (char*)d_ws;
  float*              scores = (float*)             (ws + take((size_t)NIMG * SCORE_STRIDE * 4));
  int*                sel    = (int*)               (ws + take((size_t)NIMG * M_TOT * 4));
  float*              boxDec = (float*)             (ws + take((size_t)NIMG * M_TOT * 16));
  float*              scDec  = (float*)             (ws + take((size_t)NIMG * M_TOT * 4));
  unsigned long long* keys   = (unsigned long long*)(ws + take((size_t)NIMG * M_PAD * 8));
  float*              boxS   = (float*)             (ws + take((size_t)NIMG * M_TOT * 16));
  float*              scS    = (float*)             (ws + take((size_t)NIMG * M_TOT * 4));
  unsigned*           mask   = (unsigned*)          (ws + take((size_t)NIMG * M_TOT * NWORDS * 4));
  int*                keep   = (int*)               (ws + take((size_t)NIMG * M_TOT * 4));
  (void)ws_size; (void)n_in; (void)in_sizes;

  zero_kernel<<<(out_size + 255) / 256, 256, 0, stream>>>(out, out_size);
  score_kernel<<<(NIMG * SCORE_TOTAL + 255) / 256, 256, 0, stream>>>(
      cls[0], cls[1], cls[2], cls[3], cls[4], scores);
  topk_kernel<<<NIMG * 5, 1024, 0, stream>>>(scores, sel);
  decode_kernel<<<(NIMG * M_PAD + 255) / 256, 256, 0, stream>>>(
      reg[0], reg[1], reg[2], reg[3], reg[4], scores, sel, img_h, img_w,
      boxDec, scDec, keys);
  for (int k = 2; k <= M_PAD; k <<= 1)
    for (int j = k >> 1; j > 0; j >>= 1)
      bitonic_kernel<<<(NIMG * M_PAD) / 256, 256, 0, stream>>>(keys, k, j);
  reorder_kernel<<<(NIMG * M_TOT + 255) / 256, 256, 0, stream>>>(keys, boxDec, scDec, boxS, scS);
  {
    long long waves = (long long)NIMG * (M_TOT / 16) * NWORDS;   // 313600, /4 per block
    mask_kernel<<<(unsigned)(waves / 4), 128, 0, stream>>>(boxS, mask);
  }
  nms_scan_kernel<<<NIMG, 512, 0, stream>>>(scS, mask, keep);
  output_kernel<<<NIMG, 1024, 0, stream>>>(boxS, scS, keep, out);
}